// EEGGraphNet_84602265797129
// MI455X (gfx1250) — compile-verified
//
#include <hip/hip_runtime.h>
#include <hip/hip_bf16.h>

typedef float v2f __attribute__((ext_vector_type(2)));
typedef float v8f __attribute__((ext_vector_type(8)));

#define N_ELEC 64
#define BATCH  2048
#define BN_INVN (1.0f / 131072.0f)   // 1 / (BATCH * N_ELEC)

// ---------------------------------------------------------------------------
// Kernel 0: build dense normalized adjacency A (64x64) with self-loops.
// A[d][s] = rsqrt(deg[s]) * rsqrt(deg[d]) for each edge, A[n][n] = 1/deg[n].
// Built in LDS with exact +1.0 atomics (order-independent -> deterministic).
// ---------------------------------------------------------------------------
__global__ void build_adj_kernel(const int* __restrict__ src,
                                 const int* __restrict__ dst,
                                 int E, float* __restrict__ Ag) {
    __shared__ float deg[N_ELEC];
    __shared__ float As[N_ELEC * N_ELEC];
    const int t = threadIdx.x;
    if (t < N_ELEC) deg[t] = 1.0f;
    for (int i = t; i < N_ELEC * N_ELEC; i += 256) As[i] = 0.0f;
    __syncthreads();
    for (int e = t; e < E; e += 256) atomicAdd(&deg[dst[e]], 1.0f);
    __syncthreads();
    // Each directed edge (s,d) is unique -> plain stores, no races.
    for (int e = t; e < E; e += 256) {
        int s = src[e], d = dst[e];
        As[d * N_ELEC + s] = rsqrtf(deg[s]) * rsqrtf(deg[d]);
    }
    if (t < N_ELEC) As[t * N_ELEC + t] = 1.0f / deg[t];
    __syncthreads();
    for (int i = t; i < N_ELEC * N_ELEC; i += 256) Ag[i] = As[i];
}

// ---------------------------------------------------------------------------
// BN finalize: fold 2048 deterministic per-block partials into scale/shift.
//   y = scale[c]*x + shift[c]  ==  g*(x-m)*rsqrt(v+eps)+be
// ---------------------------------------------------------------------------
__global__ void bn_finalize_kernel(const float* __restrict__ psum,
                                   const float* __restrict__ psumsq,
                                   const float* __restrict__ g,
                                   const float* __restrict__ be,
                                   int C, int nblocks,
                                   float* __restrict__ scale,
                                   float* __restrict__ shift) {
    int c = threadIdx.x;
    if (c >= C) return;
    float s = 0.0f, q = 0.0f;
    for (int b = 0; b < nblocks; ++b) {
        s += psum[(size_t)b * C + c];
        q += psumsq[(size_t)b * C + c];
    }
    float m  = s * BN_INVN;
    float v  = q * BN_INVN - m * m;
    float sc = g[c] * rsqrtf(v + 1e-5f);
    scale[c] = sc;
    shift[c] = be[c] - m * sc;
}

// ---------------------------------------------------------------------------
// GCN layer: one block per batch element. H = A * (act(X) * W) + b
// MODE 0: act = identity, write H + stats
// MODE 1: act = relu(bn(.)), write H + stats
// MODE 2: act = relu(bn(.)), fuse mean-pool over nodes -> out (B, FOUT)
// All matmuls via V_WMMA_F32_16X16X4_F32. 256 threads = 8 wave32.
// ---------------------------------------------------------------------------
template <int FIN, int FOUT, int MODE>
__global__ __launch_bounds__(256)
void gcn_layer_kernel(const float* __restrict__ Xg,     // (BATCH, 64, FIN)
                      const float* __restrict__ Wg,     // (FIN, FOUT)
                      const float* __restrict__ bias,   // (FOUT)
                      const float* __restrict__ Ag,     // (64, 64)
                      const float* __restrict__ scale,  // (FIN) MODE!=0
                      const float* __restrict__ shift,  // (FIN) MODE!=0
                      float* __restrict__ Hg,           // (BATCH,64,FOUT) MODE<2
                      float* __restrict__ psum,         // (BATCH,FOUT)    MODE<2
                      float* __restrict__ psumsq,       // (BATCH,FOUT)    MODE<2
                      float* __restrict__ outg)         // (BATCH,FOUT)    MODE==2
{
    constexpr int LDA = N_ELEC + 4;   // pads: ld % 64 == 4 -> conflict-free
    constexpr int LDX = FIN + 4;
    constexpr int LDW = FOUT + 4;
    constexpr int LDT = FOUT + 4;
    constexpr int NT    = FOUT / 16;  // column tiles (4 or 8)
    constexpr int TPW   = (4 * NT) / 8; // tiles per wave (2 or 4)
    constexpr int MSTEP = 8 / NT;       // row-tile stride per wave

    __shared__ float sA[N_ELEC * LDA];
    __shared__ float sX[N_ELEC * LDX];
    __shared__ float sW[FIN * LDW];
    __shared__ float sT[N_ELEC * LDT];
    __shared__ float sScale[FIN];
    __shared__ float sShift[FIN];

    const int tid  = threadIdx.x;
    const int b    = blockIdx.x;
    const int lane = tid & 31;
    const int wave = tid >> 5;
    const int hv   = lane >> 4;   // half-wave selects K pair / row group
    const int li   = lane & 15;   // M (A-operand) or N (B-operand) index

    const float* xb = Xg + (size_t)b * N_ELEC * FIN;

    // ---- stage LDS ----
    for (int i = tid; i < N_ELEC * N_ELEC; i += 256)
        sA[(i >> 6) * LDA + (i & 63)] = Ag[i];
    for (int i = tid; i < FIN * FOUT; i += 256)
        sW[(i / FOUT) * LDW + (i % FOUT)] = Wg[i];
    if (MODE != 0) {
        for (int c = tid; c < FIN; c += 256) { sScale[c] = scale[c]; sShift[c] = shift[c]; }
    } else {
        for (int i = tid; i < N_ELEC * FIN; i += 256)
            sX[(i / FIN) * LDX + (i % FIN)] = xb[i];
    }
    __syncthreads();
    if (MODE != 0) {   // fused BN + ReLU on the previous layer's output
        for (int i = tid; i < N_ELEC * FIN; i += 256) {
            int c = i % FIN;
            float v = fmaf(sScale[c], xb[i], sShift[c]);
            sX[(i / FIN) * LDX + c] = fmaxf(v, 0.0f);
        }
        __syncthreads();
    }

    const int no    = wave % NT;      // this wave's fixed column tile
    const int mbase = wave / NT;

    // ---- T = X * W  (K = FIN) ----
    v8f acc[TPW];
    v8f vzero = {};
#pragma unroll
    for (int t = 0; t < TPW; ++t) acc[t] = vzero;
    for (int k = 0; k < FIN; k += 4) {
        const int ka = k + hv * 2;
        v2f bv;
        bv.x = sW[ka * LDW + no * 16 + li];
        bv.y = sW[(ka + 1) * LDW + no * 16 + li];
#pragma unroll
        for (int t = 0; t < TPW; ++t) {
            const int mo = mbase + t * MSTEP;
            v2f av = *(const v2f*)&sX[(mo * 16 + li) * LDX + ka];
            acc[t] = __builtin_amdgcn_wmma_f32_16x16x4_f32(
                false, av, false, bv, (short)0, acc[t], false, false);
        }
    }
#pragma unroll
    for (int t = 0; t < TPW; ++t) {
        const int mo = mbase + t * MSTEP;
#pragma unroll
        for (int v = 0; v < 8; ++v)
            sT[(mo * 16 + v + 8 * hv) * LDT + no * 16 + li] = acc[t][v];
    }
    __syncthreads();

    // ---- H = A * T + b  (K = 64) ----
#pragma unroll
    for (int t = 0; t < TPW; ++t) acc[t] = vzero;
    for (int k = 0; k < N_ELEC; k += 4) {
        const int ka = k + hv * 2;
        v2f bv;
        bv.x = sT[ka * LDT + no * 16 + li];
        bv.y = sT[(ka + 1) * LDT + no * 16 + li];
#pragma unroll
        for (int t = 0; t < TPW; ++t) {
            const int mo = mbase + t * MSTEP;
            v2f av = *(const v2f*)&sA[(mo * 16 + li) * LDA + ka];
            acc[t] = __builtin_amdgcn_wmma_f32_16x16x4_f32(
                false, av, false, bv, (short)0, acc[t], false, false);
        }
    }
    const float bb = bias[no * 16 + li];
    __syncthreads();                       // everyone done reading sT
#pragma unroll
    for (int t = 0; t < TPW; ++t) {
        const int mo = mbase + t * MSTEP;
#pragma unroll
        for (int v = 0; v < 8; ++v)
            sT[(mo * 16 + v + 8 * hv) * LDT + no * 16 + li] = acc[t][v] + bb;
    }
    __syncthreads();

    // ---- epilogue ----
    if (MODE < 2) {
        float* hb = Hg + (size_t)b * N_ELEC * FOUT;
        for (int i = tid; i < N_ELEC * FOUT; i += 256)
            hb[i] = sT[(i / FOUT) * LDT + (i % FOUT)];
        if (tid < FOUT) {  // deterministic per-block channel stats
            float s = 0.0f, q = 0.0f;
            for (int m = 0; m < N_ELEC; ++m) {
                float h = sT[m * LDT + tid];
                s += h; q += h * h;
            }
            psum[(size_t)b * FOUT + tid]   = s;
            psumsq[(size_t)b * FOUT + tid] = q;
        }
    } else {
        if (tid < FOUT) {  // mean pool over electrodes
            float s = 0.0f;
            for (int m = 0; m < N_ELEC; ++m) s += sT[m * LDT + tid];
            outg[(size_t)b * FOUT + tid] = s * (1.0f / 64.0f);
        }
    }
}

// ---------------------------------------------------------------------------
extern "C" void kernel_launch(void* const* d_in, const int* in_sizes, int n_in,
                              void* d_out, int out_size, void* d_ws, size_t ws_size,
                              hipStream_t stream) {
    const float* x   = (const float*)d_in[0];   // (2048, 64, 64)
    const float* W1  = (const float*)d_in[1];   // (64, 64)
    const float* b1  = (const float*)d_in[2];   // (64)
    const float* W2  = (const float*)d_in[3];   // (64, 128)
    const float* b2  = (const float*)d_in[4];   // (128)
    const float* W3  = (const float*)d_in[5];   // (128, 128)
    const float* b3  = (const float*)d_in[6];   // (128)
    const float* g1  = (const float*)d_in[7];
    const float* be1 = (const float*)d_in[8];
    const float* g2  = (const float*)d_in[9];
    const float* be2 = (const float*)d_in[10];
    const int*   src = (const int*)d_in[11];
    const int*   dst = (const int*)d_in[12];
    const int    E   = in_sizes[11];

    float* ws = (float*)d_ws;
    size_t off = 0;
    float* A      = ws + off; off += 64 * 64;
    float* scale1 = ws + off; off += 64;
    float* shift1 = ws + off; off += 64;
    float* scale2 = ws + off; off += 128;
    float* shift2 = ws + off; off += 128;
    float* H1     = ws + off; off += (size_t)BATCH * 64 * 64;
    float* H2     = ws + off; off += (size_t)BATCH * 64 * 128;
    float* ps1    = ws + off; off += (size_t)BATCH * 64;
    float* pq1    = ws + off; off += (size_t)BATCH * 64;
    float* ps2    = ws + off; off += (size_t)BATCH * 128;
    float* pq2    = ws + off; off += (size_t)BATCH * 128;

    build_adj_kernel<<<1, 256, 0, stream>>>(src, dst, E, A);

    gcn_layer_kernel<64, 64, 0><<<BATCH, 256, 0, stream>>>(
        x, W1, b1, A, nullptr, nullptr, H1, ps1, pq1, nullptr);
    bn_finalize_kernel<<<1, 128, 0, stream>>>(ps1, pq1, g1, be1, 64, BATCH, scale1, shift1);

    gcn_layer_kernel<64, 128, 1><<<BATCH, 256, 0, stream>>>(
        H1, W2, b2, A, scale1, shift1, H2, ps2, pq2, nullptr);
    bn_finalize_kernel<<<1, 128, 0, stream>>>(ps2, pq2, g2, be2, 128, BATCH, scale2, shift2);

    gcn_layer_kernel<128, 128, 2><<<BATCH, 256, 0, stream>>>(
        H2, W3, b3, A, scale2, shift2, nullptr, nullptr, nullptr, (float*)d_out);
}